// NeutrinoGravNetWithRegression_3556232921277
// MI455X (gfx1250) — compile-verified
//
#include <hip/hip_runtime.h>
#include <math.h>

#define NB    64
#define NPG   1024
#define NPTS  (NB*NPG)
#define KNN   16

typedef __attribute__((ext_vector_type(2))) float v2f;
typedef __attribute__((ext_vector_type(8))) float v8f;

struct BlkParams {
  const float *W1,*b1,*W2,*b2,*W3,*b3,*sW,*sb,*hW,*hb,*oW,*ob;
  const float *xin; int xin_stride; int fin;
  float *xf; float *prebn; float *bnacc;
};

// 16x16 (K=16) GEMM tile via 4 chained V_WMMA_F32_16X16X4_F32.
// tile: A operand, row-major [16][16] in LDS. Wp: B operand [16][16] in LDS.
// Lane roles: mn = lane&15, kb = (lane>>4)*2.
//   A-frag chunk q: row=mn, k = q*4+kb, q*4+kb+1
//   B-frag chunk q: col=mn, same k
__device__ __forceinline__ v8f gemm16(const float tile[][16],
                                      const float* __restrict__ Wp,
                                      int mn, int kb) {
  v8f c = {0.f,0.f,0.f,0.f,0.f,0.f,0.f,0.f};
#pragma unroll
  for (int q = 0; q < 4; ++q) {
    const int k0 = q*4 + kb;
    v2f a, bf;
    a.x  = tile[mn][k0];
    a.y  = tile[mn][k0+1];
    bf.x = Wp[k0*16 + mn];
    bf.y = Wp[(k0+1)*16 + mn];
    c = __builtin_amdgcn_wmma_f32_16x16x4_f32(false, a, false, bf,
                                              (short)0, c, false, false);
  }
  return c;
}

// ---------------- pool0: gm = mean(x per graph); x0 = [x, gm] ----------------
__global__ __launch_bounds__(256) void k_pool0(const float* __restrict__ x,
                                               float* __restrict__ x0) {
  const int b = blockIdx.x, tid = threadIdx.x;
  __shared__ float part[256][3];
  __shared__ float gm[3];
  float s0 = 0.f, s1 = 0.f, s2 = 0.f;
  for (int r = tid; r < NPG; r += 256) {
    const float* xp = x + ((size_t)b*NPG + r)*3;
    s0 += xp[0]; s1 += xp[1]; s2 += xp[2];
  }
  part[tid][0] = s0; part[tid][1] = s1; part[tid][2] = s2;
  __syncthreads();
  if (tid < 3) {
    float s = 0.f;
    for (int i = 0; i < 256; ++i) s += part[i][tid];
    gm[tid] = s * (1.f/NPG);
  }
  __syncthreads();
  for (int r = tid; r < NPG; r += 256) {
    size_t gi = (size_t)b*NPG + r;
    for (int c = 0; c < 3; ++c) {
      x0[gi*6 + c]     = x[gi*3 + c];
      x0[gi*6 + 3 + c] = gm[c];
    }
  }
}

// ---------------- zero the BN accumulator (32 floats) ----------------
__global__ void k_zero(float* acc) {
  if (threadIdx.x < 32) acc[threadIdx.x] = 0.f;
}

// ---------------- main per-graph block kernel ----------------
__global__ __launch_bounds__(256) void k_block(BlkParams p) {
  const int b = blockIdx.x, tid = threadIdx.x;
  const int lane = tid & 31, wave = tid >> 5;
  const int half = lane >> 4, mn = lane & 15, kb = half * 2;

  __shared__ float s_lds[NPG][4];        // s0,s1,s2,|s|^2        16 KB
  __shared__ float h_lds[NPG][16];       //                       64 KB
  __shared__ float d2t[8][16][16];       // per-wave tile          8 KB
  __shared__ int   nidx[8][16][16];      //                        8 KB
  __shared__ float nwgt[8][16][16];      //                        8 KB
  __shared__ float aggt[8][16][32];      // mean|max              16 KB
  __shared__ float wW1p[256], wb1[16], wW2p[256], wb2[16], wW3p[256], wb3[16];
  __shared__ float wsWp[256], wsb[4], whWp[256], whb[16], woW[768], wob[16];
  __shared__ float bnsum[16], bnsq[16];

  // stage weights into LDS (zero-padded to 16x16 where needed)
  for (int i = tid; i < 256; i += 256) {
    const int k = i >> 4, n = i & 15;
    wW1p[i] = (k < p.fin) ? p.W1[k*16 + n] : 0.f;   // (fin,16) row-major, fo=16
    wW2p[i] = p.W2[i];
    wW3p[i] = p.W3[i];
    whWp[i] = p.hW[i];
    wsWp[i] = (n < 3) ? p.sW[k*3 + n] : 0.f;        // (16,3) padded to 16 cols
  }
  for (int i = tid; i < 768; i += 256) woW[i] = p.oW[i];
  if (tid < 16) {
    wb1[tid]=p.b1[tid]; wb2[tid]=p.b2[tid]; wb3[tid]=p.b3[tid];
    whb[tid]=p.hb[tid]; wob[tid]=p.ob[tid];
    bnsum[tid]=0.f; bnsq[tid]=0.f;
  }
  if (tid < 3) wsb[tid] = p.sb[tid];
  __syncthreads();

  float (*tile)[16] = (float (*)[16])d2t[wave];   // per-wave 16x16 staging tile

  // ---- phase 0: per-tile MLP + s/h projections, all via WMMA f32 16x16x4 ----
  for (int j = 0; j < 8; ++j) {
    const int r0 = (wave*8 + j) * 16;
    // stage input tile (zero-pad cols >= fin); lane writes col mn, rows half*8..+8
#pragma unroll
    for (int v = 0; v < 8; ++v) {
      const int m = v + half*8;
      const size_t gi = (size_t)b*NPG + r0 + m;
      tile[m][mn] = (mn < p.fin) ? p.xin[gi*(size_t)p.xin_stride + mn] : 0.f;
    }
    __builtin_amdgcn_wave_barrier();

    // ft1 + ELU
    {
      v8f c = gemm16(tile, wW1p, mn, kb);
#pragma unroll
      for (int v = 0; v < 8; ++v) {
        const int m = v + half*8;
        float a = c[v] + wb1[mn];
        tile[m][mn] = a > 0.f ? a : (expf(a) - 1.f);
      }
      __builtin_amdgcn_wave_barrier();
    }
    // ft2 + ELU
    {
      v8f c = gemm16(tile, wW2p, mn, kb);
#pragma unroll
      for (int v = 0; v < 8; ++v) {
        const int m = v + half*8;
        float a = c[v] + wb2[mn];
        tile[m][mn] = a > 0.f ? a : (expf(a) - 1.f);
      }
      __builtin_amdgcn_wave_barrier();
    }
    // ft3 + tanh -> xf (tile + global)
    {
      v8f c = gemm16(tile, wW3p, mn, kb);
#pragma unroll
      for (int v = 0; v < 8; ++v) {
        const int m = v + half*8;
        const size_t gi = (size_t)b*NPG + r0 + m;
        float a = tanhf(c[v] + wb3[mn]);
        tile[m][mn] = a;
        p.xf[gi*16 + mn] = a;
      }
      __builtin_amdgcn_wave_barrier();
    }
    // s projection (cols 0..2 valid) -> s_lds, then |s|^2
    {
      v8f c = gemm16(tile, wsWp, mn, kb);
      if (mn < 3) {
#pragma unroll
        for (int v = 0; v < 8; ++v) {
          const int m = v + half*8;
          s_lds[r0 + m][mn] = c[v] + wsb[mn];
        }
      }
      __builtin_amdgcn_wave_barrier();
      if (lane < 16) {
        const int r = r0 + lane;
        const float a0 = s_lds[r][0], a1 = s_lds[r][1], a2 = s_lds[r][2];
        s_lds[r][3] = a0*a0 + a1*a1 + a2*a2;
      }
      __builtin_amdgcn_wave_barrier();
    }
    // h projection -> h_lds  (tile still holds xf)
    {
      v8f c = gemm16(tile, whWp, mn, kb);
#pragma unroll
      for (int v = 0; v < 8; ++v) {
        const int m = v + half*8;
        h_lds[r0 + m][mn] = c[v] + whb[mn];
      }
      __builtin_amdgcn_wave_barrier();
    }
  }
  __threadfence_block();
  __syncthreads();

  // ---- phase 1: Gram via WMMA -> d2 -> register top-16 ----
  for (int j = 0; j < 8; ++j) {
    const int r0 = (wave*8 + j) * 16;
    v2f afr;                                    // A: 16x4, k=3 zero-padded
    afr.x = s_lds[r0 + mn][kb];
    afr.y = half ? 0.f : s_lds[r0 + mn][1];

    float bd[KNN]; int bi[KNN];
#pragma unroll
    for (int n = 0; n < KNN; ++n) { bd[n] = 3.4e38f; bi[n] = 0; }

    for (int ct = 0; ct < 64; ++ct) {
      const int c0 = ct * 16;
      v2f bfr;
      bfr.x = s_lds[c0 + mn][kb];
      bfr.y = half ? 0.f : s_lds[c0 + mn][1];
      v8f cz = {0.f,0.f,0.f,0.f,0.f,0.f,0.f,0.f};
      v8f g = __builtin_amdgcn_wmma_f32_16x16x4_f32(false, afr, false, bfr,
                                                    (short)0, cz, false, false);
      const float sqc = s_lds[c0 + mn][3];
#pragma unroll
      for (int v = 0; v < 8; ++v) {
        const int m = v + half*8;
        float d2 = s_lds[r0 + m][3] + sqc - 2.f * g[v];
        tile[m][mn] = fmaxf(d2, 0.f);
      }
      __builtin_amdgcn_wave_barrier();
      if (lane < 16) {
#pragma unroll
        for (int n = 0; n < 16; ++n) {
          const float cd = tile[lane][n];
          if (cd < bd[KNN-1]) {
            bd[KNN-1] = cd; bi[KNN-1] = c0 + n;
#pragma unroll
            for (int t = KNN-1; t > 0; --t) {
              if (bd[t] < bd[t-1]) {
                float td = bd[t]; bd[t] = bd[t-1]; bd[t-1] = td;
                int   ti = bi[t]; bi[t] = bi[t-1]; bi[t-1] = ti;
              }
            }
          }
        }
      }
      __builtin_amdgcn_wave_barrier();
    }

    // publish knn (idx, w = exp(-10 d2))
    if (lane < 16) {
#pragma unroll
      for (int n = 0; n < KNN; ++n) {
        nidx[wave][lane][n] = bi[n];
        nwgt[wave][lane][n] = expf(-10.f * bd[n]);
      }
    }
    __builtin_amdgcn_wave_barrier();

    // ---- phase 2: gather, weighted mean + max over 16 neighbors ----
    for (int mq = 0; mq < 8; ++mq) {
      const int m = half*8 + mq;
      float sm = 0.f, mx = -3.4e38f;
#pragma unroll
      for (int n = 0; n < KNN; ++n) {
        const int   jj = nidx[wave][m][n];
        const float wj = nwgt[wave][m][n];
        const float mv = h_lds[jj][mn] * wj;
        sm += mv; mx = fmaxf(mx, mv);
      }
      aggt[wave][m][mn]      = sm * (1.f/KNN);
      aggt[wave][m][16 + mn] = mx;
    }
    __builtin_amdgcn_wave_barrier();

    // ---- phase 3: [xf | mean | max](48) @ oW(48x16) via 12 chained WMMAs ----
    {
      const size_t rowbase = ((size_t)b*NPG + r0 + mn) * 16;   // A row = mn
      v8f c = {0.f,0.f,0.f,0.f,0.f,0.f,0.f,0.f};
#pragma unroll
      for (int q = 0; q < 12; ++q) {
        const int k0 = q*4 + kb;
        v2f a, bf;
        if (q < 4) {                    // k in [0,16): xf from global
          a.x = p.xf[rowbase + k0];
          a.y = p.xf[rowbase + k0 + 1];
        } else {                        // k in [16,48): agg from LDS
          a.x = aggt[wave][mn][k0 - 16];
          a.y = aggt[wave][mn][k0 - 15];
        }
        bf.x = woW[k0*16 + mn];
        bf.y = woW[(k0+1)*16 + mn];
        c = __builtin_amdgcn_wmma_f32_16x16x4_f32(false, a, false, bf,
                                                  (short)0, c, false, false);
      }
#pragma unroll
      for (int v = 0; v < 8; ++v) {
        const int m = v + half*8;
        const size_t gi = (size_t)b*NPG + r0 + m;
        const float acc = c[v] + wob[mn];
        p.prebn[gi*16 + mn] = acc;
        atomicAdd(&bnsum[mn], acc);
        atomicAdd(&bnsq[mn],  acc*acc);
      }
    }
    __builtin_amdgcn_wave_barrier();
  }

  __syncthreads();
  if (tid < 16) {
    atomicAdd(&p.bnacc[tid],      bnsum[tid]);
    atomicAdd(&p.bnacc[16 + tid], bnsq[tid]);
  }
}

// ---------------- batch-norm apply -> xc column slice ----------------
__global__ __launch_bounds__(256) void k_bn(const float* __restrict__ prebn,
                                            const float* __restrict__ bnacc,
                                            const float* __restrict__ gma,
                                            const float* __restrict__ bta,
                                            float* __restrict__ xcslice) {
  const int idx = blockIdx.x * 256 + threadIdx.x;   // over NPTS*16
  const int gi = idx >> 4, f = idx & 15;
  const float mu  = bnacc[f] * (1.f/NPTS);
  const float var = bnacc[16+f] * (1.f/NPTS) - mu*mu;
  const float v = gma[f] * (prebn[idx] - mu) * rsqrtf(var + 1e-5f) + bta[f];
  xcslice[(size_t)gi*48 + f] = v;
}

// ---------------- per-graph pooling + classifier head ----------------
__global__ __launch_bounds__(64) void k_head(const float* __restrict__ xc,
                                             const float* __restrict__ W1, const float* __restrict__ b1,
                                             const float* __restrict__ W2, const float* __restrict__ b2,
                                             const float* __restrict__ W3, const float* __restrict__ b3,
                                             float* __restrict__ out) {
  const int b = blockIdx.x, t = threadIdx.x;
  __shared__ float pool[48], h1[64], h2[32];
  if (t < 48) {
    float s = 0.f;
    for (int r = 0; r < NPG; ++r) {
      if (r + 64 < NPG)
        __builtin_prefetch(xc + ((size_t)b*NPG + r + 64)*48 + t, 0, 0);
      s += xc[((size_t)b*NPG + r)*48 + t];
    }
    pool[t] = s * (1.f/NPG);
  }
  __syncthreads();
  {
    float a = b1[t];
    for (int i = 0; i < 48; ++i) a += pool[i]*W1[i*64+t];
    h1[t] = fmaxf(a, 0.f);
  }
  __syncthreads();
  if (t < 32) {
    float a = b2[t];
    for (int i = 0; i < 64; ++i) a += h1[i]*W2[i*32+t];
    h2[t] = fmaxf(a, 0.f);
  }
  __syncthreads();
  if (t < 3) {
    float a = b3[t];
    for (int i = 0; i < 32; ++i) a += h2[i]*W3[i*3+t];
    out[b*3 + t] = a;
  }
}

extern "C" void kernel_launch(void* const* d_in, const int* in_sizes, int n_in,
                              void* d_out, int out_size, void* d_ws, size_t ws_size,
                              hipStream_t stream) {
  const float* x = (const float*)d_in[0];
  // d_in[1] = batch (implicit: contiguous 1024-point graphs)
  auto P = [&](int i) { return (const float*)d_in[i]; };
  static const int base[3] = {2, 16, 30};   // per-block param start (14 arrays each)

  float* ws    = (float*)d_ws;
  float* x0    = ws;                          // NPTS*6
  float* xf    = x0    + (size_t)NPTS*6;      // NPTS*16
  float* prebn = xf    + (size_t)NPTS*16;     // NPTS*16
  float* xc    = prebn + (size_t)NPTS*16;     // NPTS*48
  float* bnacc = xc    + (size_t)NPTS*48;     // 32

  k_pool0<<<NB, 256, 0, stream>>>(x, x0);

  for (int blk = 0; blk < 3; ++blk) {
    const int pb = base[blk];
    BlkParams bp;
    bp.W1 = P(pb+0);  bp.b1 = P(pb+1);
    bp.W2 = P(pb+2);  bp.b2 = P(pb+3);
    bp.W3 = P(pb+4);  bp.b3 = P(pb+5);
    bp.sW = P(pb+6);  bp.sb = P(pb+7);
    bp.hW = P(pb+8);  bp.hb = P(pb+9);
    bp.oW = P(pb+10); bp.ob = P(pb+11);
    const float* bng = P(pb+12);
    const float* bnb = P(pb+13);
    bp.fin        = (blk == 0) ? 6 : 16;
    bp.xin        = (blk == 0) ? x0 : (xc + (size_t)(blk-1)*16);
    bp.xin_stride = (blk == 0) ? 6 : 48;
    bp.xf = xf; bp.prebn = prebn; bp.bnacc = bnacc;

    k_zero <<<1, 32, 0, stream>>>(bnacc);
    k_block<<<NB, 256, 0, stream>>>(bp);
    k_bn   <<<(NPTS*16)/256, 256, 0, stream>>>(prebn, bnacc, bng, bnb, xc + (size_t)blk*16);
  }

  k_head<<<NB, 64, 0, stream>>>(xc, P(44), P(45), P(46), P(47), P(48), P(49),
                                (float*)d_out);
}